// Corr1d_4750233830058
// MI455X (gfx1250) — compile-verified
//
#include <hip/hip_runtime.h>
#include <hip/hip_bf16.h>

typedef __attribute__((ext_vector_type(2))) float v2f;
typedef __attribute__((ext_vector_type(8))) float v8f;

// Problem constants (fixed by the reference)
constexpr int BN   = 8;
constexpr int C    = 128;
constexpr int H    = 160;
constexpr int Wd   = 320;
constexpr int DISP = 64;
constexpr int HW   = H * Wd;        // 51200 elements per channel plane
constexpr int CHW  = C * HW;        // elements per batch image

constexpr int CHUNK   = 32;         // channels staged in LDS per pass
constexpr int NPAIR   = CHUNK / 2;  // 16 channel pairs per chunk
constexpr int PADCOLS = 384;        // 64 zero cols + 320 data cols
// Interleaved layout: s_fR[pair][col][2]; pair-row stride in floats.
// 800 ≡ 32 (mod 64): lanes 16..31 (pair+1) land on the other 32 banks.
constexpr int PSTRIDE = 800;        // >= 2*PADCOLS, mult of 4, ≡32 mod 64
constexpr int OUT_STRIDE = 324;     // 320 + pad, multiple of 4 for float4 I/O
constexpr int SMEM_FR  = NPAIR * PSTRIDE;      // floats
constexpr int SMEM_OUT = DISP * OUT_STRIDE;    // floats
constexpr int NWAVES   = Wd / 16;   // 20 w-tiles -> 20 waves
constexpr int BLOCK    = NWAVES * 32;          // 640 threads

__global__ __launch_bounds__(BLOCK)
void corr1d_wmma(const float* __restrict__ fL,
                 const float* __restrict__ fR,
                 float* __restrict__ out) {
  extern __shared__ float smem[];
  float* s_fR  = smem;             // [NPAIR][PADCOLS][2] with row stride PSTRIDE
  float* s_out = smem + SMEM_FR;   // [DISP][OUT_STRIDE]

  const int b  = blockIdx.x / H;
  const int hh = blockIdx.x % H;
  const float* fL_row = fL + (size_t)b * CHW + hh * Wd;
  const float* fR_row = fR + (size_t)b * CHW + hh * Wd;

  const int tid  = threadIdx.x;
  const int lane = tid & 31;
  const int hlf  = lane >> 4;       // half-wave index
  const int ln   = lane & 15;       // m for A-frag, n for B/D-frags
  const int kb   = hlf * 2;         // k sub-offset held by this half-wave
  const int Wt   = (tid >> 5) * 16; // this wave's w-tile base

  v8f acc[5] = {};                  // 5 tile pairs: W' = Wt - 64 + 16*j

  for (int chunk = 0; chunk < C / CHUNK; ++chunk) {
    if (chunk) __syncthreads();
    const int cbase = chunk * CHUNK;

    // Cooperative stage: two channel rows -> interleaved pairs in LDS,
    // left-padded with 64 zero columns (kills boundary branches later).
    for (int q = tid; q < NPAIR * (PADCOLS / 4); q += BLOCK) {
      const int p    = q / (PADCOLS / 4);
      const int col4 = (q % (PADCOLS / 4)) * 4;
      float4 va, vb;
      if (col4 >= 64) {
        const float* g0 = fR_row + (size_t)(cbase + 2 * p) * HW + (col4 - 64);
        va = *(const float4*)g0;
        vb = *(const float4*)(g0 + HW);
        if (chunk + 1 < C / CHUNK) {
          // Pull next chunk's lines toward L2 while we compute this one.
          __builtin_prefetch(g0 + (size_t)CHUNK * HW, 0, 0);
          __builtin_prefetch(g0 + (size_t)(CHUNK + 1) * HW, 0, 0);
        }
      } else {
        va = make_float4(0.f, 0.f, 0.f, 0.f);
        vb = va;
      }
      float* dst = s_fR + p * PSTRIDE + col4 * 2;
      *(float4*)(dst)     = make_float4(va.x, vb.x, va.y, vb.y);
      *(float4*)(dst + 4) = make_float4(va.z, vb.z, va.w, vb.w);
    }
    __syncthreads();

    // K-loop: A from global (each fL element read exactly once),
    // B fragments as single aligned ds_load_b64 (k,k+1 interleaved).
    const float* bbase = s_fR + hlf * PSTRIDE + (Wt + ln) * 2;
#pragma unroll
    for (int r = 0; r < CHUNK; r += 4) {
      const int cg = cbase + r + kb;
      v2f a;
      a.x = fL_row[(size_t)cg * HW + Wt + ln];
      a.y = fL_row[(size_t)(cg + 1) * HW + Wt + ln];
      const float* bp = bbase + (r / 2) * PSTRIDE;
#pragma unroll
      for (int j = 0; j < 5; ++j) {
        v2f bb = *(const v2f*)(bp + 32 * j);
        // (neg_a, A, neg_b, B, c_mod, C, reuse_a, reuse_b)
        acc[j] = __builtin_amdgcn_wmma_f32_16x16x4_f32(
            false, a, false, bb, (short)0, acc[j], false, false);
      }
    }
  }

  // Scatter D tiles into LDS as [disparity][w].
  // D[m][n] with m = v + 8*hlf, n = ln; disparity i = (64 - 16j) + m - n.
#pragma unroll
  for (int j = 0; j < 5; ++j) {
#pragma unroll
    for (int v = 0; v < 8; ++v) {
      const int m = v + 8 * hlf;
      const int i = (64 - 16 * j) + m - ln;
      if (i >= 0 && i < DISP) {
        s_out[i * OUT_STRIDE + Wt + m] = acc[j][v];
      }
    }
  }
  __syncthreads();

  // Coalesced float4 store of the [DISP][Wd] slab for this (b, h).
  float* out_row = out + (size_t)b * DISP * HW + hh * Wd;
  for (int q = tid; q < DISP * (Wd / 4); q += BLOCK) {
    const int i  = q / (Wd / 4);
    const int w4 = (q % (Wd / 4)) * 4;
    float4 v = *(const float4*)(s_out + i * OUT_STRIDE + w4);
    *(float4*)(out_row + (size_t)i * HW + w4) = v;
  }
}

extern "C" void kernel_launch(void* const* d_in, const int* in_sizes, int n_in,
                              void* d_out, int out_size, void* d_ws, size_t ws_size,
                              hipStream_t stream) {
  const float* fL = (const float*)d_in[0];
  const float* fR = (const float*)d_in[1];
  float* out = (float*)d_out;

  const size_t shmem = (size_t)(SMEM_FR + SMEM_OUT) * sizeof(float); // ~131 KB
  (void)hipFuncSetAttribute((const void*)corr1d_wmma,
                            hipFuncAttributeMaxDynamicSharedMemorySize,
                            (int)shmem);
  corr1d_wmma<<<dim3(BN * H), dim3(BLOCK), shmem, stream>>>(fL, fR, out);
}